// GraphVAE_12695923327676
// MI455X (gfx1250) — compile-verified
//
#include <hip/hip_runtime.h>
#include <hip/hip_bf16.h>
#include <math.h>

typedef float v2f __attribute__((ext_vector_type(2)));
typedef float v8f __attribute__((ext_vector_type(8)));

#define NN      100000
#define EE      3200000
#define IN_DIM  256
#define HID     64
#define LAT     16
#define ENC2    32

// -------------------- degree / normalization --------------------
__global__ void gvae_deg_init(float* deg, int n) {
    int i = blockIdx.x * blockDim.x + threadIdx.x;
    if (i < n) deg[i] = 1.0f;   // self-loop contributes 1
}

__global__ void gvae_deg_accum(const int* __restrict__ dst, float* deg, int e) {
    int i = blockIdx.x * blockDim.x + threadIdx.x;
    if (i < e) atomicAdd(&deg[dst[i]], 1.0f);
}

__global__ void gvae_deg_rsqrt(float* deg, int n) {
    int i = blockIdx.x * blockDim.x + threadIdx.x;
    if (i < n) deg[i] = rsqrtf(deg[i]);
}

// -------------------- generic fp32 WMMA GEMM --------------------
// C[M x NCOLS] = act(A[M x K] @ B[K x NCOLS] + bias), all row-major.
// One wave computes one 16x16 tile using V_WMMA_F32_16X16X4_F32.
// K and NCOLS are compile-time so all loads use immediate offsets from
// two per-lane base pointers and the K loop fully unrolls into dense
// load+WMMA bursts. M, K, NCOLS are multiples of 16 -> EXEC all-1s.
// ACT: 0 = none, 1 = relu, 2 = sigmoid.
template <int ACT, int K, int NCOLS, bool HAS_BIAS>
__global__ void gvae_gemm_wmma(const float* __restrict__ A,
                               const float* __restrict__ B,
                               const float* __restrict__ bias,
                               float* __restrict__ C) {
    const int lane    = threadIdx.x & 31;
    const int wave    = threadIdx.x >> 5;
    const int colTile = blockIdx.y * 4 + wave;
    if (colTile * 16 >= NCOLS) return;          // wave-uniform guard

    const int row0 = blockIdx.x * 16;
    const int col0 = colTile * 16;
    const int half = lane >> 4;                 // 0: lanes 0-15, 1: lanes 16-31
    const int r    = lane & 15;

    // Per-lane base pointers; everything else is an immediate offset.
    const float* abase = A + (size_t)(row0 + r) * K + half * 2;          // A[row r][half*2]
    const float* bbase = B + (size_t)(half * 2) * NCOLS + col0 + r;      // B[half*2][col r]

    v8f acc = {0.f, 0.f, 0.f, 0.f, 0.f, 0.f, 0.f, 0.f};

#pragma unroll
    for (int k0 = 0; k0 < K; k0 += 4) {
        // A 16x4: lanes 0-15 hold (K=k0,k0+1) of row r; lanes 16-31 (k0+2,k0+3).
        v2f a = *(const v2f*)(abase + k0);                // contiguous pair -> b64 load
        // B 4x16: VGPR0 = rows k0 / k0+2, VGPR1 = rows k0+1 / k0+3, striped over cols.
        v2f b;
        b.x = bbase[(size_t)k0 * NCOLS];
        b.y = bbase[(size_t)(k0 + 1) * NCOLS];
        acc = __builtin_amdgcn_wmma_f32_16x16x4_f32(
                  false, a, false, b, (short)0, acc, false, false);
    }

    // D 16x16: VGPR v -> row (v + half*8), col r.
    const int c = col0 + r;
    float bv = HAS_BIAS ? bias[c] : 0.0f;
    float* crow = C + (size_t)(row0 + half * 8) * NCOLS + c;
#pragma unroll
    for (int v = 0; v < 8; ++v) {
        float val = acc[v] + bv;
        if (ACT == 1) val = fmaxf(val, 0.0f);
        if (ACT == 2) val = 1.0f / (1.0f + expf(-val));
        crow[(size_t)v * NCOLS] = val;
    }
}

// -------------------- GCN aggregation --------------------
// acc[i,c] = h[i,c] * dinv[i]^2   (self-loop message)
template <int C>
__global__ void gvae_agg_init(const float* __restrict__ h,
                              const float* __restrict__ dinv,
                              float* __restrict__ acc, int n) {
    int i = blockIdx.x * blockDim.x + threadIdx.x;
    if (i < n * C) {
        float di = dinv[i / C];                 // C is a power of two -> shift
        acc[i] = h[i] * di * di;
    }
}

// scatter-add: acc[dst] += h[src] * dinv[src]*dinv[dst], 32 channels/thread
template <int C>
__global__ void gvae_edge_scatter(const float* __restrict__ h,
                                  const int* __restrict__ src,
                                  const int* __restrict__ dst,
                                  const float* __restrict__ dinv,
                                  float* __restrict__ acc, int e) {
    constexpr int CHUNKS = C >> 5;              // 32-channel chunks per edge
    long long t = (long long)blockIdx.x * blockDim.x + threadIdx.x;
    if (t >= (long long)e * CHUNKS) return;
    int edge = (int)(t / CHUNKS);
    int c0   = (int)(t % CHUNKS) * 32;
    // prefetch upcoming edge-index entries (gfx1250 global_prefetch_b8)
    if (edge + 2048 < e) {
        __builtin_prefetch(src + edge + 2048, 0, 1);
        __builtin_prefetch(dst + edge + 2048, 0, 1);
    }
    int s = src[edge], d = dst[edge];
    float norm = dinv[s] * dinv[d];
    const float4* hs = (const float4*)(h + (size_t)s * C + c0);
    float* ad = acc + (size_t)d * C + c0;
#pragma unroll
    for (int j = 0; j < 8; ++j) {
        float4 v = hs[j];
        atomicAdd(ad + j * 4 + 0, v.x * norm);
        atomicAdd(ad + j * 4 + 1, v.y * norm);
        atomicAdd(ad + j * 4 + 2, v.z * norm);
        atomicAdd(ad + j * 4 + 3, v.w * norm);
    }
}

template <int C>
__global__ void gvae_finalize_relu(float* acc, const float* __restrict__ bias,
                                   int n) {
    int i = blockIdx.x * blockDim.x + threadIdx.x;
    if (i < n * C) acc[i] = fmaxf(acc[i] + bias[i & (C - 1)], 0.0f);
}

// enc finalize + split mu/log_var + reparameterize
__global__ void gvae_reparam(const float* __restrict__ acc2,
                             const float* __restrict__ b2,
                             const float* __restrict__ eps,
                             float* __restrict__ mu_out,
                             float* __restrict__ lv_out,
                             float* __restrict__ z, int n) {
    int i = blockIdx.x * blockDim.x + threadIdx.x;
    if (i >= n * LAT) return;
    int node = i >> 4, c = i & (LAT - 1);
    float mu = acc2[(size_t)node * ENC2 + c]       + b2[c];
    float lv = acc2[(size_t)node * ENC2 + LAT + c] + b2[LAT + c];
    mu_out[i] = mu;
    lv_out[i] = lv;
    z[i] = mu + eps[i] * expf(0.5f * lv);
}

// -------------------- launch --------------------
extern "C" void kernel_launch(void* const* d_in, const int* in_sizes, int n_in,
                              void* d_out, int out_size, void* d_ws, size_t ws_size,
                              hipStream_t stream) {
    const float* x    = (const float*)d_in[0];
    const int*   ei   = (const int*)  d_in[1];   // [2, E]: src then dst
    const float* W1   = (const float*)d_in[2];
    const float* b1   = (const float*)d_in[3];
    const float* W2   = (const float*)d_in[4];
    const float* b2   = (const float*)d_in[5];
    const float* Wd1  = (const float*)d_in[6];
    const float* bd1  = (const float*)d_in[7];
    const float* Wd2  = (const float*)d_in[8];
    const float* bd2  = (const float*)d_in[9];
    const float* eps  = (const float*)d_in[10];

    const int* src = ei;
    const int* dst = ei + EE;

    float* out_dec = (float*)d_out;                       // [N, 256]
    float* out_mu  = out_dec + (size_t)NN * IN_DIM;       // [N, 16]
    float* out_lv  = out_mu  + (size_t)NN * LAT;          // [N, 16]

    // workspace layout (floats): N*(1 + 64 + 64 + 32 + 32 + 16) ~ 84 MB
    float* dinv = (float*)d_ws;
    float* hx   = dinv + NN;                 // x@W1  [N,64]  (reused as dmat)
    float* acc1 = hx   + (size_t)NN * HID;   // agg1 -> h [N,64]
    float* encp = acc1 + (size_t)NN * HID;   // h@W2  [N,32]
    float* acc2 = encp + (size_t)NN * ENC2;  // agg2  [N,32]
    float* z    = acc2 + (size_t)NN * ENC2;  // latent [N,16]
    float* dmat = hx;                        // decoder hidden [N,64], reuse

    const int TB = 256;
    const int rowTiles = NN / 16;            // 6250, exact

    // 1) symmetric normalization
    gvae_deg_init  <<<(NN + TB - 1) / TB, TB, 0, stream>>>(dinv, NN);
    gvae_deg_accum <<<(EE + TB - 1) / TB, TB, 0, stream>>>(dst, dinv, EE);
    gvae_deg_rsqrt <<<(NN + TB - 1) / TB, TB, 0, stream>>>(dinv, NN);

    // 2) layer 1: hx = x @ W1 (WMMA f32), aggregate, +b1, relu
    gvae_gemm_wmma<0, IN_DIM, HID, false>
        <<<dim3(rowTiles, 1), 128, 0, stream>>>(x, W1, nullptr, hx);
    gvae_agg_init<HID>
        <<<((long long)NN * HID + TB - 1) / TB, TB, 0, stream>>>(hx, dinv, acc1, NN);
    gvae_edge_scatter<HID>
        <<<((long long)EE * 2 + TB - 1) / TB, TB, 0, stream>>>(hx, src, dst, dinv, acc1, EE);
    gvae_finalize_relu<HID>
        <<<((long long)NN * HID + TB - 1) / TB, TB, 0, stream>>>(acc1, b1, NN);

    // 3) layer 2: enc = h @ W2, aggregate, +b2, split, reparameterize
    gvae_gemm_wmma<0, HID, ENC2, false>
        <<<dim3(rowTiles, 1), 128, 0, stream>>>(acc1, W2, nullptr, encp);
    gvae_agg_init<ENC2>
        <<<((long long)NN * ENC2 + TB - 1) / TB, TB, 0, stream>>>(encp, dinv, acc2, NN);
    gvae_edge_scatter<ENC2>
        <<<((long long)EE * 1 + TB - 1) / TB, TB, 0, stream>>>(encp, src, dst, dinv, acc2, EE);
    gvae_reparam
        <<<((long long)NN * LAT + TB - 1) / TB, TB, 0, stream>>>(acc2, b2, eps, out_mu, out_lv, z, NN);

    // 4) decoder: d = relu(z@Wd1 + bd1); decoded = sigmoid(d@Wd2 + bd2)
    gvae_gemm_wmma<1, LAT, HID, true>
        <<<dim3(rowTiles, 1), 128, 0, stream>>>(z, Wd1, bd1, dmat);
    gvae_gemm_wmma<2, HID, IN_DIM, true>
        <<<dim3(rowTiles, 4), 128, 0, stream>>>(dmat, Wd2, bd2, out_dec);
}